// Differentiable_DRRs_18889266168097
// MI455X (gfx1250) — compile-verified
//
#include <hip/hip_runtime.h>
#include <math.h>

// ---- problem constants (from reference) ----
#define HH 256
#define WW 256
#define NSAMP 256
__device__ const float kPX     = 1.6875f;
__device__ const float kPIERCE = 216.0f;
__device__ const float kSAD    = 742.5f;
__device__ const float kDAD    = 517.15f;

typedef float v2f __attribute__((ext_vector_type(2)));
typedef float v8f __attribute__((ext_vector_type(8)));

// slab test in "sample index" units: p(i) = s + i*d, box (-1, 256) per axis
__device__ __forceinline__ void slab_axis(float s, float d, float& lo, float& hi) {
    float invd = 1.0f / d;                 // +-inf OK for d==0
    float t0 = (-1.0f  - s) * invd;
    float t1 = (256.0f - s) * invd;
    lo = fmaxf(lo, fminf(t0, t1));
    hi = fminf(hi, fmaxf(t0, t1));
}

// One block = one image row (256 pixels, 8 waves). Each wave computes its 32
// pixel target positions with two V_WMMA_F32_16X16X4_F32 (the reference's
// detector-plane einsum), then ray-marches with early-exit slab bounds.
__global__ __launch_bounds__(256) void drr_raymarch(
    const float* __restrict__ vol,     // 256^3 f32
    const float* __restrict__ batch,   // (B,6)
    float* __restrict__ img)           // (B,256,256) raw line integrals
{
    const int tid  = threadIdx.x;
    const int lane = tid & 31;
    const int wave = tid >> 5;
    const int row  = blockIdx.x;       // 0 .. B*H-1
    const int b    = row >> 8;
    const int y    = row & 255;
    const int x    = (wave << 5) | lane;

    // ---- pose (uniform per block; scalar-friendly) ----
    const float a0 = batch[b*6+0], a1 = batch[b*6+1], a2 = batch[b*6+2];
    const float tx = batch[b*6+3], ty = batch[b*6+4], tz = batch[b*6+5];
    float ca = cosf(a0), sa = sinf(a0);
    float cb = cosf(a1), sb = sinf(a1);
    float cg = cosf(a2), sg = sinf(a2);
    // R = Rz@Ry@Rx ; columns:
    float ux = ca*cb,              uy = sa*cb,              uz = -sb;       // R[:,0]
    float vx = -sa*cg + ca*sb*sg,  vy = ca*cg + sa*sb*sg,   vz = cb*sg;     // R[:,1]
    float wx = sa*sg + ca*sb*cg,   wy = -ca*sg + sa*sb*cg,  wz = cb*cg;     // R[:,2]
    float isox = 128.0f + tx, isoy = 128.0f + ty, isoz = 128.0f + tz;
    float srcx = isox + kSAD*wx, srcy = isoy + kSAD*wy, srcz = isoz + kSAD*wz;
    float dcx  = isox - kDAD*wx, dcy  = isoy - kDAD*wy, dcz  = isoz - kDAD*wz;

    // ---- WMMA FP32 16x16x4: D[m][c] = u_m*u_ax[c] + v*v_ax[c] + 1*det_c[c]
    // A (16x4): lanes 0-15 hold K=0,1 = (u_m, v); lanes 16-31 hold K=2,3 = (1, 0)
    // B (4x16): lanes 0-15 hold K=0,1 rows (u_ax, v_ax); lanes 16-31 K=2,3 (det_c, 0)
    const int n  = lane & 15;
    const int hi = lane >> 4;
    float vcoord = (float)y * kPX - kPIERCE;
    float u_low  = (float)((wave << 5) + n)      * kPX - kPIERCE;  // pixels 0..15 of wave
    float u_high = (float)((wave << 5) + 16 + n) * kPX - kPIERCE;  // pixels 16..31

    v2f A0, A1, Bv;
    A0.x = hi ? 1.0f : u_low;   A0.y = hi ? 0.0f : vcoord;
    A1.x = hi ? 1.0f : u_high;  A1.y = hi ? 0.0f : vcoord;
    float uax = (n==0)?ux:((n==1)?uy:((n==2)?uz:0.0f));
    float vax = (n==0)?vx:((n==1)?vy:((n==2)?vz:0.0f));
    float dcc = (n==0)?dcx:((n==1)?dcy:((n==2)?dcz:0.0f));
    Bv.x = hi ? dcc  : uax;
    Bv.y = hi ? 0.0f : vax;

    v8f Z = {};
    v8f D0 = __builtin_amdgcn_wmma_f32_16x16x4_f32(false, A0, false, Bv, (short)0, Z, false, false);
    v8f D1 = __builtin_amdgcn_wmma_f32_16x16x4_f32(false, A1, false, Bv, (short)0, Z, false, false);

    // Scatter D columns 0..2 (held in lanes {0,1,2,16,17,18}) to owning lanes via LDS.
    __shared__ float tgt[8][32][4];
    if (n < 3) {
        int mh = hi * 8;
        #pragma unroll
        for (int r = 0; r < 8; ++r) {
            tgt[wave][mh + r][n]      = D0[r];
            tgt[wave][16 + mh + r][n] = D1[r];
        }
    }
    __syncthreads();
    float tpx = tgt[wave][lane][0];
    float tpy = tgt[wave][lane][1];
    float tpz = tgt[wave][lane][2];

    // ---- ray march ----
    float rx = tpx - srcx, ry = tpy - srcy, rz = tpz - srcz;
    float length = sqrtf(rx*rx + ry*ry + rz*rz);
    const float inv255 = 1.0f / 255.0f;          // linspace(0,1,256) spacing
    float dxs = rx * inv255, dys = ry * inv255, dzs = rz * inv255;

    float lo = 0.0f, hif = 255.0f;
    slab_axis(srcx, dxs, lo, hif);
    slab_axis(srcy, dys, lo, hif);
    slab_axis(srcz, dzs, lo, hif);
    int i0 = max((int)ceilf(lo), 0);
    int i1 = min((int)floorf(hif), 255);

    float px = srcx + (float)i0 * dxs;
    float py = srcy + (float)i0 * dys;
    float pz = srcz + (float)i0 * dzs;
    // approx per-step linear offset for prefetch (elements)
    int stepOff = ((int)dxs << 16) + ((int)dys << 8) + (int)dzs;

    float acc = 0.0f;
    for (int i = i0; i <= i1; ++i) {
        float fxf = floorf(px), fyf = floorf(py), fzf = floorf(pz);
        int ix = (int)fxf, iy = (int)fyf, iz = (int)fzf;
        float fx = px - fxf, fy = py - fyf, fz = pz - fzf;
        // mode='constant': zero weight for out-of-range neighbors (p in (-1,256))
        float wx1 = (ix <= 254) ? fx : 0.0f;   float wx0 = (ix >= 0) ? (1.0f - fx) : 0.0f;
        float wy1 = (iy <= 254) ? fy : 0.0f;   float wy0 = (iy >= 0) ? (1.0f - fy) : 0.0f;
        float wz1 = (iz <= 254) ? fz : 0.0f;   float wz0 = (iz >= 0) ? (1.0f - fz) : 0.0f;
        int ox0 = max(ix, 0) << 16, ox1 = min(ix + 1, 255) << 16;
        int oy0 = max(iy, 0) << 8,  oy1 = min(iy + 1, 255) << 8;
        int oz0 = max(iz, 0),       oz1 = min(iz + 1, 255);

        float v000 = vol[ox0 + oy0 + oz0], v001 = vol[ox0 + oy0 + oz1];
        float v010 = vol[ox0 + oy1 + oz0], v011 = vol[ox0 + oy1 + oz1];
        float v100 = vol[ox1 + oy0 + oz0], v101 = vol[ox1 + oy0 + oz1];
        float v110 = vol[ox1 + oy1 + oz0], v111 = vol[ox1 + oy1 + oz1];

        // prefetch the neighborhood two steps ahead (masked into the 2^24-elem volume)
        int pf = (ox0 + oy0 + oz0 + (stepOff << 1)) & 0x00FFFFFF;
        __builtin_prefetch((const void*)(vol + pf), 0, 0);

        float c00 = v000*wx0 + v100*wx1;
        float c01 = v001*wx0 + v101*wx1;
        float c10 = v010*wx0 + v110*wx1;
        float c11 = v011*wx0 + v111*wx1;
        float c0  = c00*wy0 + c10*wy1;
        float c1  = c01*wy0 + c11*wy1;
        acc += c0*wz0 + c1*wz1;

        px += dxs; py += dys; pz += dzs;
    }
    img[(b << 16) | (y << 8) | x] = acc * length * (1.0f / 256.0f);
}

// One block per batch image: min/max over 65536 values.
__global__ __launch_bounds__(1024) void drr_minmax(const float* __restrict__ img,
                                                   float* __restrict__ mm)
{
    const int b = blockIdx.x;
    const float* p = img + (b << 16);
    float mn = 3.0e38f, mx = -3.0e38f;
    for (int i = threadIdx.x; i < 65536; i += 1024) {
        float v = p[i];
        mn = fminf(mn, v);
        mx = fmaxf(mx, v);
    }
    __shared__ float smn[1024], smx[1024];
    smn[threadIdx.x] = mn; smx[threadIdx.x] = mx;
    __syncthreads();
    for (int s = 512; s > 0; s >>= 1) {
        if (threadIdx.x < s) {
            smn[threadIdx.x] = fminf(smn[threadIdx.x], smn[threadIdx.x + s]);
            smx[threadIdx.x] = fmaxf(smx[threadIdx.x], smx[threadIdx.x + s]);
        }
        __syncthreads();
    }
    if (threadIdx.x == 0) { mm[b*2] = smn[0]; mm[b*2+1] = smx[0]; }
}

// In-place normalize + invert.
__global__ __launch_bounds__(256) void drr_norm(float* __restrict__ img,
                                                const float* __restrict__ mm)
{
    int idx = blockIdx.x * 256 + threadIdx.x;
    int b = idx >> 16;
    float mn = mm[b*2], mx = mm[b*2+1];
    float v = img[idx];
    img[idx] = 1.0f - (v - mn) / (mx - mn);
}

extern "C" void kernel_launch(void* const* d_in, const int* in_sizes, int n_in,
                              void* d_out, int out_size, void* d_ws, size_t ws_size,
                              hipStream_t stream)
{
    (void)in_sizes; (void)n_in; (void)out_size; (void)ws_size;
    const float* vol   = (const float*)d_in[0];   // (256,256,256) f32
    const float* batch = (const float*)d_in[1];   // (2,6) f32
    // d_in[2] = n_samples (==256), baked into the kernel
    float* out = (float*)d_out;                   // (2,256,256) f32; used as raw img then normalized in place
    float* mm  = (float*)d_ws;                    // 4 floats: {mn0,mx0,mn1,mx1}

    drr_raymarch<<<2 * HH, 256, 0, stream>>>(vol, batch, out);
    drr_minmax  <<<2, 1024, 0, stream>>>(out, mm);
    drr_norm    <<<2 * HH, 256, 0, stream>>>(out, mm);
}